// PointPillarScatter_6451040878696
// MI455X (gfx1250) — compile-verified
//
#include <hip/hip_runtime.h>

// PointPillarScatter for MI455X (gfx1250).
// Pure bandwidth problem: ~0.5 GB moved, 0 FLOPs -> roofline ~22us at 23.3 TB/s.
// WMMA does not apply (no matrix math); the optimal instruction mix is
// vmem-dominated with gfx1250 cache-control (TH_*_NT) on the streaming paths.
//
// Pass 1: stream-zero the 439 MB output. Block-uniform guard -> scalar branch,
//         then 4 unconditional back-to-back nontemporal global_store_b128
//         sharing one address VGPR (imm offsets 0/4096/8192/12288 B). NT keeps
//         the 439 MB stream from thrashing the 192 MB L2.
// Pass 2: scatter. 16 lanes share one pillar -> coalesced NT b128 feature loads;
//         4 channel stores per thread use one address + immediate offsets
//         (S*4 = 857088 B fits the signed 24-bit VMEM ioffset).

typedef float v4f __attribute__((ext_vector_type(4)));

#define BEV_NX 432
#define BEV_NY 496
#define BEV_C  64
#define BEV_S  (BEV_NX * BEV_NY)   // 214272 cells per (batch, channel) plane

__global__ __launch_bounds__(256)
void pps_zero_fill(v4f* __restrict__ out, int n4) {
    // Each block covers 1024 float4s: thread t stores at base, +256, +512, +768.
    int blk = (int)blockIdx.x;
    int base = blk * 1024 + (int)threadIdx.x;
    v4f z = (v4f)0.0f;
    // Block-uniform guard: scalar compare + branch, no exec masking.
    if ((blk + 1) * 1024 <= n4) {
        // Common path: 4 unconditional coalesced NT b128 stores.
        __builtin_nontemporal_store(z, out + base);
        __builtin_nontemporal_store(z, out + base + 256);
        __builtin_nontemporal_store(z, out + base + 512);
        __builtin_nontemporal_store(z, out + base + 768);
    } else {
        // Tail block only (none when n4 % 1024 == 0, as here).
        for (int k = 0; k < 4; ++k) {
            int i = base + k * 256;
            if (i < n4) __builtin_nontemporal_store(z, out + i);
        }
    }
}

__global__ __launch_bounds__(256)
void pps_scatter(const v4f* __restrict__ feat4,     // [P][16] float4 view of [P][64]
                 const int* __restrict__ cb,
                 const int* __restrict__ cz,
                 const int* __restrict__ cy,
                 const int* __restrict__ cx,
                 float* __restrict__ out,           // [B][64][S]
                 int P) {
    int i = blockIdx.x * blockDim.x + threadIdx.x;  // over P*16
    int p = i >> 4;                                 // pillar id (16 lanes share it)
    int q = i & 15;                                 // which float4 of the 64 channels
    if (p >= P) return;

    // Coord loads: 16 consecutive lanes read the same addresses -> cache broadcast.
    int b   = cb[p];
    int idx = cz[p] + cy[p] * BEV_NX + cx[p];       // flat spatial index, nz==1

    // Coalesced 512 B/wave streaming read of features (read exactly once -> NT).
    v4f v = __builtin_nontemporal_load(feat4 + p * 16 + q);

    // Channel stride S: four b32 stores off one base with immediate offsets.
    float* base = out + (size_t)(b * BEV_C + q * 4) * BEV_S + idx;
    base[0 * BEV_S] = v.x;
    base[1 * BEV_S] = v.y;
    base[2 * BEV_S] = v.z;
    base[3 * BEV_S] = v.w;
}

extern "C" void kernel_launch(void* const* d_in, const int* in_sizes, int n_in,
                              void* d_out, int out_size, void* d_ws, size_t ws_size,
                              hipStream_t stream) {
    const v4f* feat4 = (const v4f*)d_in[0];   // pillar_features [P,64] f32
    const int* cb    = (const int*)d_in[1];   // coords_b [P]
    const int* cz    = (const int*)d_in[2];   // coords_z [P]
    const int* cy    = (const int*)d_in[3];   // coords_y [P]
    const int* cx    = (const int*)d_in[4];   // coords_x [P]
    float* out = (float*)d_out;

    const int P  = in_sizes[1];               // 128000
    const int n4 = out_size >> 2;             // out_size divisible by 4 (C=64)

    // Pass 1: zero fill (439 MB of NT b128 stores, 1024 float4s per block).
    int zb = (n4 + 1023) / 1024;
    pps_zero_fill<<<zb, 256, 0, stream>>>((v4f*)out, n4);

    // Pass 2: scatter (same stream -> ordered after the fill).
    int total = P * 16;
    int sb = (total + 255) / 256;
    pps_scatter<<<sb, 256, 0, stream>>>(feat4, cb, cz, cy, cx, out, P);
}